// CSB_2680059592815
// MI455X (gfx1250) — compile-verified
//
#include <hip/hip_runtime.h>

// ---------------------------------------------------------------------------
// CSB fused pipeline for MI455X (gfx1250), fp32 end-to-end via
// V_WMMA_F32_16X16X4_F32 (wave32, A/B = 2 VGPRs f32, C/D = 8 VGPRs f32).
// Problem: B=4, N=16384, C=384, D=192, M=N/16=1024, H=W=128.
// LDS tiles keep the unit-stride axis along K so most fragments are one
// aligned ds_load_b64 (sp-GEMM B via frag_b: two strided b32). Global->LDS staging uses the CDNA5 async-to-LDS path
// (GLOBAL_LOAD_ASYNC_TO_LDS_*, ASYNCcnt) so chunk transfers overlap WMMA
// compute with no VGPR round trip.
// ---------------------------------------------------------------------------

typedef __attribute__((ext_vector_type(2))) float v2f;
typedef __attribute__((ext_vector_type(8))) float v8f;

#define BATCH 4
#define NTOK  16384
#define CH    384
#define DH    192
#define MM    1024   // N/16 conv output positions per batch

#if defined(__gfx1250__) && __has_builtin(__builtin_amdgcn_global_load_async_to_lds_b32)
#define ASYNC_LDS 1
typedef int v4i32 __attribute__((vector_size(16)));
typedef __attribute__((address_space(1))) int*   gp32;
typedef __attribute__((address_space(3))) int*   lp32;
typedef __attribute__((address_space(1))) v4i32* gp128;
typedef __attribute__((address_space(3))) v4i32* lp128;
#else
#define ASYNC_LDS 0
#endif

// global -> LDS, one dword, per-lane LDS scatter allowed
__device__ __forceinline__ void cp_b32(float* l, const float* g) {
#if ASYNC_LDS
  __builtin_amdgcn_global_load_async_to_lds_b32((gp32)g, (lp32)l, 0, 0);
#else
  *l = *g;
#endif
}
// global -> LDS, 16 bytes (both sides 16B aligned)
__device__ __forceinline__ void cp_b128(float* l, const float* g) {
#if ASYNC_LDS
  __builtin_amdgcn_global_load_async_to_lds_b128((gp128)g, (lp128)l, 0, 0);
#else
  *(float4*)l = *(const float4*)g;
#endif
}
// LDS -> global, one dword
__device__ __forceinline__ void st_b32(float* g, const float* l) {
#if ASYNC_LDS
  __builtin_amdgcn_global_store_async_from_lds_b32((gp32)g, (lp32)l, 0, 0);
#else
  *g = *l;
#endif
}
__device__ __forceinline__ void async_wait0() {
#if ASYNC_LDS
#if __has_builtin(__builtin_amdgcn_s_wait_asynccnt)
  __builtin_amdgcn_s_wait_asynccnt(0);
#else
  asm volatile("s_wait_asynccnt 0" ::: "memory");
#endif
#endif
}

__device__ __forceinline__ v8f wmma4(v2f a, v2f b, v8f c) {
  return __builtin_amdgcn_wmma_f32_16x16x4_f32(
      false, a, false, b, (short)0, c, false, false);
}

// A 16x4 (MxK) f32: lanes 0-15 -> K{0,1}; lanes 16-31 -> K{2,3}. Contiguous pair.
__device__ __forceinline__ v2f frag_a(const float* A, int lda, int m0, int k0, int lane) {
  int m = m0 + (lane & 15);
  int k = k0 + ((lane >> 4) << 1);
  v2f r; r.x = A[m * lda + k]; r.y = A[m * lda + k + 1]; return r;
}
// B 4x16 (KxN) from a transposed store T[n][k]: contiguous pair -> one b64.
__device__ __forceinline__ v2f frag_bT(const float* T, int ldt, int k0, int n0, int lane) {
  int n = n0 + (lane & 15);
  int k = k0 + ((lane >> 4) << 1);
  v2f r; r.x = T[n * ldt + k]; r.y = T[n * ldt + k + 1]; return r;
}
// B 4x16 (KxN) from row-major storage Bm[k][n] (two strided b32 reads).
__device__ __forceinline__ v2f frag_b(const float* Bm, int ldb, int k0, int n0, int lane) {
  int n = n0 + (lane & 15);
  int k = k0 + ((lane >> 4) << 1);
  v2f r; r.x = Bm[k * ldb + n]; r.y = Bm[(k + 1) * ldb + n]; return r;
}

// ---------------------------------------------------------------------------
// Kernel 1: conv 4x4/s4 as GEMM  W(192x3072) * P(3072x1024) + bias.
// grid (16,3,B), block 256 (8 waves), C tile 64x64, K chunks of 16,
// double-buffered async LDS staging.  As ld=20 (16B rows -> b128 weights),
// BsT ld=18 (patch gather, b32).
// ---------------------------------------------------------------------------
__device__ __forceinline__ void conv_load_chunk(const float* __restrict__ x,
                                                const float* __restrict__ w,
                                                float* As, float* BsT,
                                                int b, int o0, int p0, int k0, int tid) {
  {                                      // 64x16 weights: one b128 per thread
    int i4 = tid * 4;
    int mo = i4 >> 4, kk = i4 & 15;
    cp_b128(&As[mo * 20 + kk], &w[(size_t)(o0 + mo) * 3072 + k0 + kk]);
  }
  for (int i = tid; i < 16 * 64; i += 256) {   // patch gather, transposed
    int kk = i >> 6, pl = i & 63;
    int k = k0 + kk;
    int ich = k >> 4, t = k & 15, r = t >> 2, cc = t & 3;
    int p = p0 + pl, ph = p >> 5, pw = p & 31;
    int n = (4 * ph + r) * 128 + 4 * pw + cc;
    cp_b32(&BsT[pl * 18 + kk], &x[((size_t)b * NTOK + n) * CH + DH + ich]);
  }
}

__global__ __launch_bounds__(256) void k_conv(const float* __restrict__ x,
                                              const float* __restrict__ w,
                                              const float* __restrict__ bias,
                                              float* __restrict__ x2c) {
  __shared__ float As[2][64 * 20];
  __shared__ float BsT[2][64 * 18];
  const int b = blockIdx.z;
  const int o0 = blockIdx.y * 64;
  const int p0 = blockIdx.x * 64;
  const int tid = threadIdx.x, lane = tid & 31, wid = tid >> 5;
  const int m0w = (wid >> 1) * 16;
  const int n0w = (wid & 1) * 32;

  v8f acc[2] = {};
  conv_load_chunk(x, w, As[0], BsT[0], b, o0, p0, 0, tid);
  for (int c = 0; c < 192; ++c) {
    const int cur = c & 1;
    async_wait0();        // this wave's chunk-cur transfers complete
    __syncthreads();      // all waves done (and prev readers of buf nxt done)
    if (c + 1 < 192)
      conv_load_chunk(x, w, As[cur ^ 1], BsT[cur ^ 1], b, o0, p0, (c + 1) * 16, tid);
    for (int kk = 0; kk < 16; kk += 4) {
      v2f a  = frag_a(As[cur], 20, m0w, kk, lane);
      v2f b0 = frag_bT(BsT[cur], 18, kk, n0w, lane);
      v2f b1 = frag_bT(BsT[cur], 18, kk, n0w + 16, lane);
      acc[0] = wmma4(a, b0, acc[0]);
      acc[1] = wmma4(a, b1, acc[1]);
    }
  }
  const int col = lane & 15, rbase = (lane >> 4) << 3;
  for (int j = 0; j < 2; ++j) {
    int p = p0 + n0w + j * 16 + col;
    for (int v = 0; v < 8; ++v) {
      int o = o0 + m0w + rbase + v;
      x2c[((size_t)b * DH + o) * MM + p] = acc[j][v] + bias[o];
    }
  }
}

// ---------------------------------------------------------------------------
// Kernel 2a: zero the S accumulator (4*192*192 floats).
// ---------------------------------------------------------------------------
__global__ void k_zero(float* __restrict__ p) {
  p[(size_t)blockIdx.x * 256 + threadIdx.x] = 0.0f;
}

// ---------------------------------------------------------------------------
// Kernel 2b: S[b,d,e] += sum_n x1[b,n,d]*x2[b,n,e]  (split-K, atomic reduce).
// grid (16 kslices, 9 de-tiles, B), C tile 64x64, Kslice 1024, chunks of 16,
// double-buffered async staging, coalesced global reads, LDS-side transpose.
// ---------------------------------------------------------------------------
__device__ __forceinline__ void att_load_chunk(const float* __restrict__ x,
                                               float* As, float* BsT,
                                               int b, int d0, int e0, int k0, int tid) {
  for (int i = tid; i < 1024; i += 256) {       // As[m][k] = x1[n=k0+kk, d0+m]
    int m = i & 63, kk = i >> 6;
    cp_b32(&As[m * 18 + kk], &x[((size_t)b * NTOK + k0 + kk) * CH + d0 + m]);
  }
  for (int i = tid; i < 1024; i += 256) {       // BsT[e][k] = x2[n=k0+kk, e0+e]
    int e = i & 63, kk = i >> 6;
    cp_b32(&BsT[e * 18 + kk], &x[((size_t)b * NTOK + k0 + kk) * CH + DH + e0 + e]);
  }
}

__global__ __launch_bounds__(256) void k_att_partial(const float* __restrict__ x,
                                                     float* __restrict__ S) {
  __shared__ float As[2][64 * 18];
  __shared__ float BsT[2][64 * 18];
  const int b = blockIdx.z;
  const int d0 = (blockIdx.y / 3) * 64;
  const int e0 = (blockIdx.y % 3) * 64;
  const int kbase = blockIdx.x * 1024;
  const int tid = threadIdx.x, lane = tid & 31, wid = tid >> 5;
  const int m0w = (wid >> 1) * 16;
  const int n0w = (wid & 1) * 32;

  v8f acc[2] = {};
  att_load_chunk(x, As[0], BsT[0], b, d0, e0, kbase, tid);
  for (int c = 0; c < 64; ++c) {
    const int cur = c & 1;
    async_wait0();
    __syncthreads();
    if (c + 1 < 64)
      att_load_chunk(x, As[cur ^ 1], BsT[cur ^ 1], b, d0, e0, kbase + (c + 1) * 16, tid);
    for (int kk = 0; kk < 16; kk += 4) {
      v2f a  = frag_a(As[cur], 18, m0w, kk, lane);
      v2f b0 = frag_bT(BsT[cur], 18, kk, n0w, lane);
      v2f b1 = frag_bT(BsT[cur], 18, kk, n0w + 16, lane);
      acc[0] = wmma4(a, b0, acc[0]);
      acc[1] = wmma4(a, b1, acc[1]);
    }
  }
  const int col = lane & 15, rbase = (lane >> 4) << 3;
  for (int j = 0; j < 2; ++j) {
    int e = e0 + n0w + j * 16 + col;
    for (int v = 0; v < 8; ++v) {
      int d = d0 + m0w + rbase + v;
      atomicAdd(&S[((size_t)b * DH + d) * DH + e], acc[j][v]);
    }
  }
}

// ---------------------------------------------------------------------------
// Kernel 3: in-place softmax of S over d (axis -2) per (b, e) column.
// ---------------------------------------------------------------------------
__global__ void k_softmax(float* __restrict__ S) {
  const int b = blockIdx.x, e = threadIdx.x;
  float mx = -3.4e38f;
  for (int d = 0; d < DH; ++d)
    mx = fmaxf(mx, S[((size_t)b * DH + d) * DH + e]);
  float sum = 0.0f;
  for (int d = 0; d < DH; ++d) {
    float v = __expf(S[((size_t)b * DH + d) * DH + e] - mx);
    S[((size_t)b * DH + d) * DH + e] = v;
    sum += v;
  }
  float inv = 1.0f / sum;
  for (int d = 0; d < DH; ++d)
    S[((size_t)b * DH + d) * DH + e] *= inv;
}

// ---------------------------------------------------------------------------
// Kernel 4: fused gate + sp.  Per 64-row n-tile, loop m in chunks of 64:
//   gate = sigmoid(x1_tile @ x2c[:,chunk])   (WMMA)
//   sp  += gate @ x2c[:,chunk]^T             (WMMA)
// Never materializes the 268 MB gate tensor. x2c chunks staged via async-to-
// LDS; the mc+64 chunk is also prefetched (global_prefetch_b8) into cache.
// grid (256, B), block 256 (8 waves), ~117 KB dynamic LDS.
// x1s ld=196 (16B rows -> b128), x2cT ld=194, gts ld=66.
// ---------------------------------------------------------------------------
__global__ __launch_bounds__(256) void k_gate_sp(const float* __restrict__ x,
                                                 const float* __restrict__ x2c,
                                                 float* __restrict__ sp) {
  extern __shared__ float smem[];
  float* x1s  = smem;                 // 64 x 196  (x1 tile, row-major over d)
  float* x2cT = x1s + 64 * 196;       // 64 x 194  (chunk transposed: [m][d])
  float* gts  = x2cT + 64 * 194;      // 64 x 66   (gate chunk, [n][m])
  const int b = blockIdx.y;
  const int n0 = blockIdx.x * 64;
  const int tid = threadIdx.x, lane = tid & 31, wid = tid >> 5;
  const int m0w = (wid >> 1) * 16;    // wave's 16 rows of the 64-row tile
  const int nb  = (wid & 1) * 6;      // 6 of 12 d-subtiles (192 cols)
  const int gn0 = (wid & 1) * 32;     // gate GEMM column base
  const int col = lane & 15, rbase = (lane >> 4) << 3;

  for (int i4 = tid * 4; i4 < 64 * 192; i4 += 1024) {   // b128 async copies
    int r = i4 / 192, d = i4 % 192;
    cp_b128(&x1s[r * 196 + d], &x[((size_t)b * NTOK + n0 + r) * CH + d]);
  }

  v8f spacc[6] = {};
  for (int mc = 0; mc < MM; mc += 64) {
    __syncthreads();  // prior sp-GEMM reads of x2cT complete
    for (int i = tid; i < 64 * 192; i += 256) {
      int m = i & 63, d = i >> 6;     // coalesced global, transposed LDS
      cp_b32(&x2cT[m * 194 + d], &x2c[((size_t)b * DH + d) * MM + mc + m]);
    }
    if (mc + 64 < MM && tid < DH)     // pull next chunk toward L2/L0
      __builtin_prefetch(&x2c[((size_t)b * DH + tid) * MM + mc + 64], 0, 1);
    async_wait0();
    __syncthreads();

    // gate GEMM: (64 x 192) @ (192 x 64), 2 C-tiles per wave
    v8f g[2] = {};
    for (int k = 0; k < DH; k += 4) {
      v2f a  = frag_a(x1s, 196, m0w, k, lane);
      v2f b0 = frag_bT(x2cT, 194, k, gn0, lane);
      v2f b1 = frag_bT(x2cT, 194, k, gn0 + 16, lane);
      g[0] = wmma4(a, b0, g[0]);
      g[1] = wmma4(a, b1, g[1]);
    }
    for (int j = 0; j < 2; ++j)
      for (int v = 0; v < 8; ++v) {
        float s = 1.0f / (1.0f + __expf(-g[j][v]));
        gts[(m0w + rbase + v) * 66 + gn0 + j * 16 + col] = s;
      }
    __syncthreads();

    // sp GEMM accumulate: (64x64 gate) @ (64x192 x2c^T), 6 tiles per wave
    for (int k = 0; k < 64; k += 4) {
      v2f a = frag_a(gts, 66, m0w, k, lane);
      for (int j = 0; j < 6; ++j) {
        v2f bj = frag_b(x2cT, 194, k, (nb + j) * 16, lane);  // B[k=m][d]
        spacc[j] = wmma4(a, bj, spacc[j]);
      }
    }
  }

  for (int j = 0; j < 6; ++j) {
    int d = (nb + j) * 16 + col;
    for (int v = 0; v < 8; ++v) {
      int r = m0w + rbase + v;
      sp[((size_t)b * NTOK + n0 + r) * DH + d] = spacc[j][v];
    }
  }
}

// ---------------------------------------------------------------------------
// Kernel 5: ch = x2 @ att  ((16384x192) @ (192x192) per batch).
// att K-chunks async-staged transposed ([e][k], b64 frags); x2 A-fragments
// straight from global (contiguous float2 per lane, read once per block).
// ---------------------------------------------------------------------------
__global__ __launch_bounds__(256) void k_ch(const float* __restrict__ x,
                                            const float* __restrict__ att,
                                            float* __restrict__ ch) {
  __shared__ float attsT[192 * 34];
  const int b = blockIdx.y;
  const int n0 = blockIdx.x * 64;
  const int tid = threadIdx.x, lane = tid & 31, wid = tid >> 5;
  const int m0w = (wid >> 1) * 16;
  const int nb  = (wid & 1) * 6;

  v8f acc[6] = {};
  for (int k0 = 0; k0 < DH; k0 += 32) {
    __syncthreads();                            // prior readers done
    for (int i = tid; i < 32 * 192; i += 256) {
      int kk = i / 192, e = i % 192;            // coalesced att row reads
      cp_b32(&attsT[e * 34 + kk], &att[((size_t)b * DH + k0 + kk) * DH + e]);
    }
    async_wait0();
    __syncthreads();
    for (int k = 0; k < 32; k += 4) {
      int m = n0 + m0w + (lane & 15);
      int kk = k0 + k + ((lane >> 4) << 1);
      const float* px = &x[((size_t)b * NTOK + m) * CH + DH + kk];
      v2f a; a.x = px[0]; a.y = px[1];
      for (int j = 0; j < 6; ++j) {
        v2f bj = frag_bT(attsT, 34, k, (nb + j) * 16, lane);
        acc[j] = wmma4(a, bj, acc[j]);
      }
    }
  }
  const int col = lane & 15, rbase = (lane >> 4) << 3;
  for (int j = 0; j < 6; ++j) {
    int e = (nb + j) * 16 + col;
    for (int v = 0; v < 8; ++v) {
      int r = m0w + rbase + v;
      ch[((size_t)b * NTOK + n0 + r) * DH + e] = acc[j][v];
    }
  }
}

// ---------------------------------------------------------------------------
// Kernel 6: cat -> LayerNorm -> @ proj_w^T + proj_b -> transposed store.
// cat staged with async b128; final transposed store uses
// GLOBAL_STORE_ASYNC_FROM_LDS_B32 (coalesced, no VGPR round trip; S_ENDPGM
// performs the implicit wait-idle). smem ld=392 (16B rows, bank-spread).
// ---------------------------------------------------------------------------
__global__ __launch_bounds__(256) void k_ln_proj(const float* __restrict__ sp,
                                                 const float* __restrict__ ch,
                                                 const float* __restrict__ lnw,
                                                 const float* __restrict__ lnb,
                                                 const float* __restrict__ pw,
                                                 const float* __restrict__ pb,
                                                 float* __restrict__ out) {
  extern __shared__ float smem[];   // 64 x 392
  const int CLD = 392;
  const int b = blockIdx.y;
  const int n0 = blockIdx.x * 64;
  const int tid = threadIdx.x, lane = tid & 31, wid = tid >> 5;

  for (int i4 = tid * 4; i4 < 64 * DH; i4 += 1024) {
    int r = i4 / DH, c = i4 % DH;
    cp_b128(&smem[r * CLD + c],      &sp[((size_t)b * NTOK + n0 + r) * DH + c]);
    cp_b128(&smem[r * CLD + DH + c], &ch[((size_t)b * NTOK + n0 + r) * DH + c]);
  }
  async_wait0();
  __syncthreads();

  if (tid < 64) {  // one row per thread: LayerNorm over 384
    float* row = smem + tid * CLD;
    float mu = 0.0f;
    for (int c = 0; c < CH; ++c) mu += row[c];
    mu *= (1.0f / CH);
    float var = 0.0f;
    for (int c = 0; c < CH; ++c) { float d = row[c] - mu; var += d * d; }
    var *= (1.0f / CH);
    float inv = rsqrtf(var + 1e-5f);
    for (int c = 0; c < CH; ++c)
      row[c] = (row[c] - mu) * inv * lnw[c] + lnb[c];
  }
  __syncthreads();

  // GEMM: (64 x 384) @ (384 x 384), out[n,o] = sum_c ln[n,c]*pw[o,c]
  const int m0w = (wid >> 1) * 16;
  const int nbase = (wid & 1) * 192;      // 12 o-subtiles per wave
  v8f acc[12] = {};
  for (int k = 0; k < CH; k += 4) {
    v2f a = frag_a(smem, CLD, m0w, k, lane);
    int kk = k + ((lane >> 4) << 1);
    for (int j = 0; j < 12; ++j) {
      int o = nbase + j * 16 + (lane & 15);
      const float* pp = &pw[(size_t)o * CH + kk];
      v2f bb; bb.x = pp[0]; bb.y = pp[1];
      acc[j] = wmma4(a, bb, acc[j]);
    }
  }
  __syncthreads();  // all cat reads complete; reuse smem as out tile [n][o]

  const int col = lane & 15, rbase = (lane >> 4) << 3;
  for (int j = 0; j < 12; ++j) {
    int o = nbase + j * 16 + col;
    for (int v = 0; v < 8; ++v)
      smem[(m0w + rbase + v) * CLD + o] = acc[j][v] + pb[o];
  }
  __syncthreads();

  for (int i = tid; i < 64 * CH; i += 256) {   // async LDS->global, coalesced
    int o = i >> 6, nl = i & 63;
    st_b32(&out[((size_t)b * CH + o) * NTOK + n0 + nl], &smem[nl * CLD + o]);
  }
}

// ---------------------------------------------------------------------------
// Host launcher
// ---------------------------------------------------------------------------
extern "C" void kernel_launch(void* const* d_in, const int* in_sizes, int n_in,
                              void* d_out, int out_size, void* d_ws, size_t ws_size,
                              hipStream_t stream) {
  const float* x      = (const float*)d_in[0];
  const float* conv_w = (const float*)d_in[1];
  const float* conv_b = (const float*)d_in[2];
  const float* ln_w   = (const float*)d_in[3];
  const float* ln_b   = (const float*)d_in[4];
  const float* proj_w = (const float*)d_in[5];
  const float* proj_b = (const float*)d_in[6];
  float* out = (float*)d_out;

  float* ws  = (float*)d_ws;
  float* x2c = ws;                   // 4*192*1024      =    786,432 f
  float* S   = x2c + 786432;         // 4*192*192       =    147,456 f (att in-place)
  float* sp  = S + 147456;           // 4*16384*192     = 12,582,912 f
  float* ch  = sp + 12582912;        // 4*16384*192     = 12,582,912 f

  k_conv<<<dim3(16, 3, BATCH), 256, 0, stream>>>(x, conv_w, conv_b, x2c);
  k_zero<<<dim3(576), 256, 0, stream>>>(S);
  k_att_partial<<<dim3(16, 9, BATCH), 256, 0, stream>>>(x, S);
  k_softmax<<<dim3(BATCH), 192, 0, stream>>>(S);

  size_t sm4 = (size_t)(64 * 196 + 64 * 194 + 64 * 66) * sizeof(float);  // ~117 KB
  (void)hipFuncSetAttribute(reinterpret_cast<const void*>(k_gate_sp),
                            hipFuncAttributeMaxDynamicSharedMemorySize, (int)sm4);
  k_gate_sp<<<dim3(256, BATCH), 256, sm4, stream>>>(x, x2c, sp);

  k_ch<<<dim3(256, BATCH), 256, 0, stream>>>(x, S, ch);

  size_t sm6 = (size_t)(64 * 392) * sizeof(float);                       // ~100 KB
  (void)hipFuncSetAttribute(reinterpret_cast<const void*>(k_ln_proj),
                            hipFuncAttributeMaxDynamicSharedMemorySize, (int)sm6);
  k_ln_proj<<<dim3(256, BATCH), 256, sm6, stream>>>(sp, ch, ln_w, ln_b,
                                                    proj_w, proj_b, out);
}